// LenetOptNet_41970420417252
// MI455X (gfx1250) — compile-verified
//
#include <hip/hip_runtime.h>
#include <math.h>

typedef __attribute__((ext_vector_type(2))) float v2f;
typedef __attribute__((ext_vector_type(8))) float v8f;

#define BQ 2048      // batch
#define NH 50        // QP dim n
#define MM 200       // constraints m
#define GS 65        // LDS row stride for G (padded cols 50..64)
#define HS 65        // LDS row stride for H / Q
#define NITER 15

// ---------------------------------------------------------------------------
// Block helpers (128-thread blocks, uniform control flow)
// ---------------------------------------------------------------------------
__device__ inline float block_sum(float v, float* red, int tid) {
    red[tid] = v; __syncthreads();
    for (int off = 64; off > 0; off >>= 1) {
        if (tid < off) red[tid] += red[tid + off];
        __syncthreads();
    }
    float r = red[0]; __syncthreads();
    return r;
}

__device__ inline float block_min(float v, float* red, int tid) {
    red[tid] = v; __syncthreads();
    for (int off = 64; off > 0; off >>= 1) {
        if (tid < off) red[tid] = fminf(red[tid], red[tid + off]);
        __syncthreads();
    }
    float r = red[0]; __syncthreads();
    return r;
}

// In-place Cholesky of leading 50x50 of Hs (row stride HS). Lower triangle valid.
__device__ inline void chol50(float* Hs, int tid) {
    for (int k = 0; k < NH; ++k) {
        if (tid == 0) Hs[k * HS + k] = sqrtf(Hs[k * HS + k]);
        __syncthreads();
        { int i = k + 1 + tid; if (i < NH) Hs[i * HS + k] /= Hs[k * HS + k]; }
        __syncthreads();
        { int j = k + 1 + tid;
          if (j < NH) {
              float ljk = Hs[j * HS + k];
              for (int i2 = j; i2 < NH; ++i2) Hs[i2 * HS + j] -= Hs[i2 * HS + k] * ljk;
          } }
        __syncthreads();
    }
}

// Solve (L L^T) x = y using Cholesky factor in Hs. yv destroyed, result -> xout.
__device__ inline void tri_solve(const float* Hs, float* yv, float* xout, int tid) {
    for (int j = 0; j < NH; ++j) {                 // forward: L u = y
        if (tid == 0) yv[j] /= Hs[j * HS + j];
        __syncthreads();
        int i = j + 1 + tid;
        if (i < NH) yv[i] -= Hs[i * HS + j] * yv[j];
        __syncthreads();
    }
    for (int j = NH - 1; j >= 0; --j) {            // backward: L^T x = u
        if (tid == 0) xout[j] = yv[j] / Hs[j * HS + j];
        __syncthreads();
        if (tid < j) yv[tid] -= Hs[j * HS + tid] * xout[j];
        __syncthreads();
    }
}

// One Newton solve of the IPM KKT system given factored H.
__device__ inline void newton_step(const float* Gs, const float* Hs,
                                   const float* rzv, const float* rp,
                                   const float* svec, const float* lamv,
                                   const float* rcv,
                                   float* wk, float* yv, float* dzv,
                                   float* dsv, float* dlv, int tid) {
    for (int k = tid; k < MM; k += 128)
        wk[k] = (lamv[k] * rp[k] - rcv[k]) / svec[k];
    __syncthreads();
    if (tid < NH) {
        float acc = -rzv[tid];
        for (int k = 0; k < MM; ++k) acc -= wk[k] * Gs[k * GS + tid];
        yv[tid] = acc;
    }
    __syncthreads();
    tri_solve(Hs, yv, dzv, tid);
    __syncthreads();
    for (int k = tid; k < MM; k += 128) {
        float acc = -rp[k];
        const float* gr = Gs + k * GS;
        for (int j = 0; j < NH; ++j) acc -= gr[j] * dzv[j];
        dsv[k] = acc;
        dlv[k] = (-rcv[k] - lamv[k] * acc) / svec[k];
    }
    __syncthreads();
}

__device__ inline float step_len(const float* sv, const float* dsw,
                                 const float* lv, const float* dlw,
                                 float* red, int tid) {
    float amin = 1e10f;
    for (int k = tid; k < MM; k += 128) {
        float dsk = dsw[k], dlk = dlw[k];
        if (dsk < 0.f) amin = fminf(amin, -sv[k] / dsk);
        if (dlk < 0.f) amin = fminf(amin, -lv[k] / dlk);
    }
    float m = block_min(amin, red, tid);
    return fminf(1.f, 0.99f * m);
}

// ---------------------------------------------------------------------------
// Kernel 0: Q = tril(L) tril(L)^T + eps I ; chol(Q).  One 128-thread block.
// ---------------------------------------------------------------------------
__global__ void prep_kernel(const float* __restrict__ L,
                            float* __restrict__ Qg, float* __restrict__ Cg) {
    __shared__ float Cs[NH * HS];
    int tid = threadIdx.x;
    for (int idx = tid; idx < NH * NH; idx += 128) {
        int i = idx / NH, j = idx % NH;
        int mn = i < j ? i : j;
        float acc = (i == j) ? 1e-4f : 0.f;
        for (int k = 0; k <= mn; ++k) acc += L[i * NH + k] * L[j * NH + k];
        Qg[idx] = acc;
        Cs[i * HS + j] = acc;
    }
    __syncthreads();
    chol50(Cs, tid);
    __syncthreads();
    for (int idx = tid; idx < NH * NH; idx += 128) {
        int i = idx / NH, j = idx % NH;
        Cg[idx] = Cs[i * HS + j];
    }
}

// ---------------------------------------------------------------------------
// Kernel 1: conv1(5x5) + maxpool2 -> F1[B][20][12][12]
// ---------------------------------------------------------------------------
__global__ void conv1_kernel(const float* __restrict__ X, const float* __restrict__ W,
                             const float* __restrict__ Bb, float* __restrict__ F1) {
    int idx = blockIdx.x * blockDim.x + threadIdx.x;
    if (idx >= BQ * 20 * 144) return;
    int b = idx / (20 * 144);
    int r = idx % (20 * 144);
    int c = r / 144;
    int yx = r % 144;
    int y = yx / 12, x = yx % 12;
    const float* xim = X + b * 784;
    const float* w = W + c * 25;
    float best = -3.4e38f;
    for (int py = 0; py < 2; ++py)
        for (int px = 0; px < 2; ++px) {
            int oy = 2 * y + py, ox = 2 * x + px;
            float acc = Bb[c];
            for (int ky = 0; ky < 5; ++ky) {
                const float* row = xim + (oy + ky) * 28 + ox;
                const float* wr = w + ky * 5;
#pragma unroll
                for (int kx = 0; kx < 5; ++kx) acc += row[kx] * wr[kx];
            }
            best = fmaxf(best, acc);
        }
    F1[idx] = best;
}

// ---------------------------------------------------------------------------
// Kernel 2: conv2(5x5, 20ch) + maxpool2 -> F[B][800].  One block per image;
// image (11.5KB) + weights (100KB) staged in LDS.
// ---------------------------------------------------------------------------
__global__ void conv2_kernel(const float* __restrict__ F1, const float* __restrict__ W2,
                             const float* __restrict__ B2, float* __restrict__ F) {
    extern __shared__ float sm2[];
    float* sf1 = sm2;          // 2880
    float* sw2 = sm2 + 2880;   // 25000
    int b = blockIdx.x, tid = threadIdx.x;
    for (int i = tid; i < 2880; i += 256) sf1[i] = F1[b * 2880 + i];
    for (int i = tid; i < 25000; i += 256) sw2[i] = W2[i];
    __syncthreads();
    for (int o = tid; o < 800; o += 256) {
        int c2 = o >> 4;
        int yx = o & 15;
        int y = yx >> 2, x = yx & 3;
        float best = -3.4e38f;
        for (int py = 0; py < 2; ++py)
            for (int px = 0; px < 2; ++px) {
                int oy = 2 * y + py, ox = 2 * x + px;
                float acc = B2[c2];
                const float* wc = sw2 + c2 * 500;
                for (int ci = 0; ci < 20; ++ci) {
                    const float* im = sf1 + ci * 144;
                    const float* wk2 = wc + ci * 25;
                    for (int ky = 0; ky < 5; ++ky) {
                        const float* row = im + (oy + ky) * 12 + ox;
#pragma unroll
                        for (int kx = 0; kx < 5; ++kx) acc += row[kx] * wk2[ky * 5 + kx];
                    }
                }
                best = fmaxf(best, acc);
            }
        F[b * 800 + o] = best;
    }
}

// ---------------------------------------------------------------------------
// Kernel 3: FC projections via WMMA f32 16x16x4:
//   P[b][0:50]=f@o_w^T+o_b, P[b][50:100]=f@z0_w^T+z0_b, P[b][100:300]=f@s0_w^T+s0_b
// Grid (B/16, 5) x 128 threads; each wave owns one 16(M)x16(N) tile, K=800.
// Out-of-range columns are CLAMPED to a valid weight row (never stored), so the
// inner loop has no divergence: straight global_load_b64 + v_wmma.
// ---------------------------------------------------------------------------
__global__ void fc_kernel(const float* __restrict__ f,
                          const float* __restrict__ ow, const float* __restrict__ ob,
                          const float* __restrict__ zw, const float* __restrict__ zb,
                          const float* __restrict__ sw, const float* __restrict__ sb,
                          float* __restrict__ P) {
    int tid = threadIdx.x, lane = tid & 31, wave = tid >> 5;
    int m0 = blockIdx.x * 16;
    int j0 = blockIdx.y * 64 + wave * 16;
    int l15 = lane & 15;
    int col = j0 + l15;
    int colc = (col < 300) ? col : 299;          // clamp: always a valid row
    int koff = (lane < 16) ? 0 : 2;
    int rbase = (lane < 16) ? 0 : 8;

    float bias;
    const float* wrow;
    if (colc < 50)       { wrow = ow + colc * 800;         bias = ob[colc]; }
    else if (colc < 100) { wrow = zw + (colc - 50) * 800;  bias = zb[colc - 50]; }
    else                 { wrow = sw + (colc - 100) * 800; bias = sb[colc - 100]; }

    v8f c;
#pragma unroll
    for (int r = 0; r < 8; ++r) c[r] = bias;

    const float* arow = f + (m0 + l15) * 800;
    for (int k = 0; k < 800; k += 4) {
        float a0 = arow[k + koff];
        float a1 = arow[k + koff + 1];
        float b0 = wrow[k + koff];
        float b1 = wrow[k + koff + 1];
        v2f A = {a0, a1};
        v2f Bv = {b0, b1};
        c = __builtin_amdgcn_wmma_f32_16x16x4_f32(false, A, false, Bv, (short)0, c,
                                                  false, false);
    }
    if (col < 300) {
#pragma unroll
        for (int r = 0; r < 8; ++r) P[(m0 + r + rbase) * 300 + col] = c[r];
    }
}

// ---------------------------------------------------------------------------
// Kernel 4: full IPM QP solve, one 128-thread block (4 waves) per batch item.
// ---------------------------------------------------------------------------
__global__ void qp_kernel(const float* __restrict__ Qg, const float* __restrict__ Cg,
                          const float* __restrict__ Gg, const float* __restrict__ P,
                          float* __restrict__ out) {
    extern __shared__ float sm[];
    float* Gs   = sm;                  // 200*65
    float* Qs   = Gs + MM * GS;        // 64*65 (zero padded)
    float* Hs   = Qs + 64 * HS;        // 64*65
    float* zv   = Hs + 64 * HS;        // 64
    float* pvec = zv + 64;             // 64
    float* rzv  = pvec + 64;           // 64
    float* yv   = rzv + 64;            // 64
    float* dzv  = yv + 64;             // 64
    float* hvec = dzv + 64;            // 200
    float* svec = hvec + MM;           // 200
    float* lamv = svec + MM;           // 200
    float* rp   = lamv + MM;           // 200
    float* dvec = rp + MM;             // 200
    float* wk   = dvec + MM;           // 200
    float* rcv  = wk + MM;             // 200
    float* dsv  = rcv + MM;            // 200
    float* dlv  = dsv + MM;            // 200
    float* red  = dlv + MM;            // 128
    float* scal = red + 128;           // 4

    int b = blockIdx.x, tid = threadIdx.x, lane = tid & 31, wave = tid >> 5;

    // ---- Stage shared matrices into LDS -----------------------------------
    for (int idx = tid; idx < MM * NH; idx += 128) {
        int k = idx / NH, j = idx % NH;
        Gs[k * GS + j] = Gg[idx];
    }
    for (int idx = tid; idx < MM * 15; idx += 128) {       // pad cols 50..64
        int k = idx / 15, j = NH + idx % 15;
        Gs[k * GS + j] = 0.f;
    }
    for (int idx = tid; idx < 64 * 64; idx += 128) {
        int i = idx >> 6, j = idx & 63;
        Qs[i * HS + j] = (i < NH && j < NH) ? Qg[i * NH + j] : 0.f;
    }
    for (int idx = tid; idx < NH * NH; idx += 128) {       // chol(Q) -> Hs for init
        int i = idx / NH, j = idx % NH;
        Hs[i * HS + j] = Cg[idx];
    }
    for (int i = tid; i < NH; i += 128) {
        pvec[i] = P[b * 300 + i];
        dzv[i]  = P[b * 300 + 50 + i];   // z0 (temp)
    }
    for (int k = tid; k < MM; k += 128)
        svec[k] = P[b * 300 + 100 + k];  // s0 (temp)
    __syncthreads();

    // h = G z0 + s0
    for (int k = tid; k < MM; k += 128) {
        float acc = svec[k];
        const float* gr = Gs + k * GS;
        for (int j = 0; j < NH; ++j) acc += gr[j] * dzv[j];
        hvec[k] = acc;
    }
    // z = Q^{-1}(-p)
    for (int i = tid; i < NH; i += 128) yv[i] = -pvec[i];
    __syncthreads();
    tri_solve(Hs, yv, zv, tid);
    for (int k = tid; k < MM; k += 128) { svec[k] = 1.f; lamv[k] = 1.f; }
    __syncthreads();

    int i0 = wave * 16;
    int l15 = lane & 15;
    int koff = (lane < 16) ? 0 : 2;
    int rbase = (lane < 16) ? 0 : 8;
    int mrow = i0 + l15;

    for (int iter = 0; iter < NITER; ++iter) {
        // residuals, d, mu --------------------------------------------------
        float musum = 0.f;
        for (int k = tid; k < MM; k += 128) {
            const float* gr = Gs + k * GS;
            float gz = 0.f;
            for (int j = 0; j < NH; ++j) gz += gr[j] * zv[j];
            rp[k]   = gz + svec[k] - hvec[k];
            dvec[k] = lamv[k] / svec[k];
            musum  += svec[k] * lamv[k];
        }
        if (tid < NH) {
            float acc = pvec[tid];
            const float* qr = Qs + tid * HS;
            for (int j = 0; j < NH; ++j) acc += qr[j] * zv[j];
            for (int k = 0; k < MM; ++k) acc += Gs[k * GS + tid] * lamv[k];
            rzv[tid] = acc;
        }
        float mu = block_sum(musum, red, tid) * (1.f / 200.f);

        // H = Q + G^T D G via WMMA f32 16x16x4 ------------------------------
        for (int jt = 0; jt < 4; ++jt) {
            int j0 = jt * 16;
            int colj = j0 + l15;
            v8f c;
#pragma unroll
            for (int r = 0; r < 8; ++r) c[r] = Qs[(i0 + rbase + r) * HS + colj];
            for (int k0 = 0; k0 < MM; k0 += 4) {
                int ka = k0 + koff;
                float a0 = Gs[ka * GS + mrow] * dvec[ka];
                float a1 = Gs[(ka + 1) * GS + mrow] * dvec[ka + 1];
                float b0 = Gs[ka * GS + colj];
                float b1 = Gs[(ka + 1) * GS + colj];
                v2f A = {a0, a1};
                v2f Bv = {b0, b1};
                c = __builtin_amdgcn_wmma_f32_16x16x4_f32(false, A, false, Bv,
                                                          (short)0, c, false, false);
            }
#pragma unroll
            for (int r = 0; r < 8; ++r) Hs[(i0 + rbase + r) * HS + colj] = c[r];
        }
        __syncthreads();

        chol50(Hs, tid);

        // predictor (affine) ------------------------------------------------
        for (int k = tid; k < MM; k += 128) rcv[k] = svec[k] * lamv[k];
        __syncthreads();
        newton_step(Gs, Hs, rzv, rp, svec, lamv, rcv, wk, yv, dzv, dsv, dlv, tid);
        float a_aff = step_len(svec, dsv, lamv, dlv, red, tid);
        float masum = 0.f;
        for (int k = tid; k < MM; k += 128)
            masum += (svec[k] + a_aff * dsv[k]) * (lamv[k] + a_aff * dlv[k]);
        float mu_aff = block_sum(masum, red, tid) * (1.f / 200.f);
        float ratio = mu_aff / (mu + 1e-12f);
        float sigma = ratio * ratio * ratio;

        // corrector (Mehrotra) ----------------------------------------------
        for (int k = tid; k < MM; k += 128)
            rcv[k] = svec[k] * lamv[k] + dsv[k] * dlv[k] - sigma * mu;
        __syncthreads();
        newton_step(Gs, Hs, rzv, rp, svec, lamv, rcv, wk, yv, dzv, dsv, dlv, tid);
        float a = step_len(svec, dsv, lamv, dlv, red, tid);

        if (tid < NH) zv[tid] += a * dzv[tid];
        for (int k = tid; k < MM; k += 128) {
            svec[k] += a * dsv[k];
            lamv[k] += a * dlv[k];
        }
        __syncthreads();
    }

    // log_softmax(z[:10]) ----------------------------------------------------
    if (tid == 0) {
        float mx = zv[0];
        for (int i = 1; i < 10; ++i) mx = fmaxf(mx, zv[i]);
        float se = 0.f;
        for (int i = 0; i < 10; ++i) se += expf(zv[i] - mx);
        scal[0] = mx;
        scal[1] = logf(se);
    }
    __syncthreads();
    if (tid < 10) out[b * 10 + tid] = zv[tid] - scal[0] - scal[1];
}

// ---------------------------------------------------------------------------
extern "C" void kernel_launch(void* const* d_in, const int* in_sizes, int n_in,
                              void* d_out, int out_size, void* d_ws, size_t ws_size,
                              hipStream_t stream) {
    const float* x  = (const float*)d_in[0];
    const float* w1 = (const float*)d_in[1];
    const float* b1 = (const float*)d_in[2];
    const float* w2 = (const float*)d_in[3];
    const float* b2 = (const float*)d_in[4];
    const float* ow = (const float*)d_in[5];
    const float* ob = (const float*)d_in[6];
    const float* zw = (const float*)d_in[7];
    const float* zb = (const float*)d_in[8];
    const float* sw = (const float*)d_in[9];
    const float* sb = (const float*)d_in[10];
    const float* L  = (const float*)d_in[11];
    const float* G  = (const float*)d_in[12];
    float* out = (float*)d_out;

    float* ws = (float*)d_ws;
    float* Qg = ws;                        // 2500
    float* Cg = Qg + 2500;                 // 2500
    float* F1 = Cg + 2500;                 // 2048*2880
    float* F  = F1 + (size_t)BQ * 2880;    // 2048*800
    float* P  = F  + (size_t)BQ * 800;     // 2048*300

    prep_kernel<<<1, 128, 0, stream>>>(L, Qg, Cg);
    conv1_kernel<<<(BQ * 20 * 144 + 255) / 256, 256, 0, stream>>>(x, w1, b1, F1);
    conv2_kernel<<<BQ, 256, (2880 + 25000) * sizeof(float), stream>>>(F1, w2, b2, F);
    fc_kernel<<<dim3(BQ / 16, 5), 128, 0, stream>>>(F, ow, ob, zw, zb, sw, sb, P);

    size_t qp_smem = (size_t)(MM * GS + 2 * 64 * HS + 5 * 64 + 9 * MM + 128 + 4)
                     * sizeof(float);     // ~94 KB, fits 320 KB/WGP LDS
    qp_kernel<<<BQ, 128, qp_smem, stream>>>(Qg, Cg, G, P, out);
}